// NaiveSSMLayer_38448547234650
// MI455X (gfx1250) — compile-verified
//
#include <hip/hip_runtime.h>
#include <hip/hip_bf16.h>

typedef __attribute__((ext_vector_type(2))) float v2f;
typedef __attribute__((ext_vector_type(8))) float v8f;
typedef __attribute__((ext_vector_type(4))) unsigned int u32x4;
typedef __attribute__((ext_vector_type(8))) int i32x8;
typedef __attribute__((ext_vector_type(4))) int i32x4;

#define NST  256
#define LSEQ 4096
#define DMOD 2048

#if defined(__has_builtin)
#if __has_builtin(__builtin_amdgcn_tensor_load_to_lds) && \
    __has_builtin(__builtin_amdgcn_s_wait_tensorcnt)
#define USE_TDM 1
#endif
#endif
#ifndef USE_TDM
#define USE_TDM 0
#endif

// ---------------- Phase 1: discretization helpers ----------------

__global__ void k_scaleA(float* __restrict__ E, const float* __restrict__ A, float h) {
    int i = blockIdx.x * blockDim.x + threadIdx.x;
    if (i < NST * NST) E[i] = h * A[i];
}

__global__ void k_initT(float* __restrict__ T, const float* __restrict__ E) {
    int i = blockIdx.x * blockDim.x + threadIdx.x;
    if (i < NST * NST) {
        int r = i >> 8, c = i & 255;
        T[i] = E[i] + (r == c ? 1.0f : 0.0f);
    }
}

__global__ void k_abar(float* __restrict__ Ab, const float* __restrict__ T) {
    int i = blockIdx.x * blockDim.x + threadIdx.x;
    if (i < NST * NST) {
        int r = i >> 8, c = i & 255;
        Ab[i] = 2.0f * T[i] - (r == c ? 1.0f : 0.0f);
    }
}

// Generic small GEMM: C[i,j] = sum_k A[i*256+k] * B[k*ldb + j]  (+I optional)
__global__ void k_gemm256(float* __restrict__ C, const float* __restrict__ A,
                          const float* __restrict__ B, int ncols, int ldb, int ldc,
                          int addI) {
    __shared__ float As[16][16];
    __shared__ float Bs[16][17];
    const int tx = threadIdx.x, ty = threadIdx.y;
    const int j = blockIdx.x * 16 + tx;
    const int i = blockIdx.y * 16 + ty;
    float acc = 0.0f;
    for (int kt = 0; kt < 16; ++kt) {
        As[ty][tx] = A[i * NST + kt * 16 + tx];
        Bs[ty][tx] = (j < ncols) ? B[(kt * 16 + ty) * (size_t)ldb + j] : 0.0f;
        __syncthreads();
#pragma unroll
        for (int k = 0; k < 16; ++k) acc += As[ty][k] * Bs[k][tx];
        __syncthreads();
    }
    if (j < ncols) C[i * (size_t)ldc + j] = acc + ((addI && i == j) ? 1.0f : 0.0f);
}

// Bbar = step * (I + E + ... + E^iters) B   (single block, 256 threads)
__global__ void k_bbar(float* __restrict__ Bbar, const float* __restrict__ E,
                       const float* __restrict__ Bv, float stepv, int iters) {
    __shared__ float xl[NST];
    const int i = threadIdx.x;
    const float bi = Bv[i];
    xl[i] = bi;
    __syncthreads();
    for (int it = 0; it < iters; ++it) {
        float s = bi;
        for (int k = 0; k < NST; ++k) s += E[i * NST + k] * xl[k];
        __syncthreads();
        xl[i] = s;
        __syncthreads();
    }
    Bbar[i] = stepv * xl[i];
}

__global__ void k_setU0(float* __restrict__ U, const float* __restrict__ Bbar) {
    int i = threadIdx.x;
    U[(size_t)i * LSEQ] = Bbar[i];
}

// K[l] = sum_n C[n] * U[n*LSEQ + l]  (coalesced across l)
__global__ void k_kvec(float* __restrict__ Kv, const float* __restrict__ Cv,
                       const float* __restrict__ U) {
    int l = blockIdx.x * blockDim.x + threadIdx.x;
    float s = 0.0f;
    for (int n = 0; n < NST; ++n) s += Cv[n] * U[(size_t)n * LSEQ + l];
    Kv[l] = s;
}

// ---------------- Phase 3: causal conv as WMMA GEMM ----------------
// out(2048x4096) = Y(2048x4096) @ T(4096x4096), T[j,t] = K[t-j] (t>=j), + D*y
// Block: 128x128 C-tile, 256 threads = 8 waves (2 M x 4 N), wave tile 64x32.
// Y tile staged via TDM (tensor_load_to_lds) with native LDS padding;
// Toeplitz B tile synthesized once per kb from K (LDS->LDS), then both
// operands feed v_wmma_f32_16x16x4_f32 via ds_load_b64 fragments.
__global__ __launch_bounds__(256) void k_conv(float* __restrict__ out,
                                              const float* __restrict__ y,
                                              const float* __restrict__ Kv,
                                              const float* __restrict__ Dp) {
    __shared__ float Yt[128][36];      // pad 36: conflict-free, = TDM pad (32+4 dwords)
    __shared__ float Btt[128][34];     // Toeplitz tile, transposed [n][k], pad 34
    __shared__ float Kls[LSEQ];        // whole kernel vector, 16 KB

    const int tid  = threadIdx.x;
    const int lane = tid & 31;
    const int wave = tid >> 5;
    const int wm   = wave & 1;         // 0..1  -> M offset 0/64
    const int wn   = wave >> 1;        // 0..3  -> N offset 0/32/64/96
    const int lhi  = lane >> 4;        // 0/1
    const int llo  = lane & 15;
    const int m0   = blockIdx.y * 128;
    const int n0   = blockIdx.x * 128;

    for (int i = tid; i < LSEQ; i += 256) Kls[i] = Kv[i];
    __syncthreads();                   // Kls ready for Btt staging

    v8f acc[4][2];
#pragma unroll
    for (int im = 0; im < 4; ++im)
#pragma unroll
        for (int in = 0; in < 2; ++in)
#pragma unroll
            for (int v = 0; v < 8; ++v) acc[im][in][v] = 0.0f;

#if !USE_TDM
    const int r0 = tid >> 3;          // 0..31
    const int c4 = (tid & 7) * 4;     // 0,4,...,28
#endif
    // Btt staging split: thread -> one row n, 16 consecutive k
    const int sn  = tid & 127;
    const int sk0 = (tid >> 7) * 16;

    for (int kb = 0; kb < LSEQ / 32; ++kb) {
#if USE_TDM
        if (wave == 0) {
            // --- Tensor DMA descriptor (D#), per CDNA5 ISA ---
            const unsigned lds_base =
                (unsigned)(unsigned long long)(size_t)&Yt[0][0];
            const unsigned long long ga =
                (unsigned long long)(size_t)(y + (size_t)m0 * LSEQ + kb * 32);
            u32x4 g0;
            g0[0] = 1u;                                   // count=1, user mode
            g0[1] = lds_base;                             // lds_addr [63:32]
            g0[2] = (unsigned)ga;                         // global_addr [95:64]
            g0[3] = ((unsigned)(ga >> 32) & 0x01FFFFFFu)  // global_addr [120:96]
                    | (2u << 30);                         // type=2 ("image")
            i32x8 g1;
            g1[0] = (int)((2u << 16)    // data_size = 4 bytes
                        | (1u << 20)    // pad_enable
                        | (4u << 22)    // pad_interval: 32 dwords
                        | (3u << 25));  // pad_amount:   4 dwords  -> LDS stride 36
            g1[1] = (int)(4096u << 16); // tensor_dim0 = 4096 (lo16)
            g1[2] = (int)(2048u << 16); // tensor_dim0 hi=0 | tensor_dim1 = 2048 (lo16)
            g1[3] = (int)(32u << 16);   // tensor_dim1 hi=0 | tile_dim0 = 32
            g1[4] = 128;                // tile_dim1 = 128 | tile_dim2 = 0
            g1[5] = 4096;               // tensor_dim0_stride (lo32)
            g1[6] = 0;                  // stride0 hi | stride1 lo
            g1[7] = 0;                  // stride1 hi
            const i32x4 gz4 = {0, 0, 0, 0};
            const i32x8 gz8 = {0, 0, 0, 0, 0, 0, 0, 0};
            // 6-arg form (clang-23 / therock-10.0 lane): extra descriptor
            // group is only consumed by >2D tensors; zero for our 2D tile.
            __builtin_amdgcn_tensor_load_to_lds(g0, g1, gz4, gz4, gz8, 0);
        }
#else
#pragma unroll
        for (int ii = 0; ii < 4; ++ii) {
            const int row = r0 + 32 * ii;
            const float4 v = *(const float4*)(y + (size_t)(m0 + row) * LSEQ + kb * 32 + c4);
            *(float4*)&Yt[row][c4] = v;
        }
#endif
        // Stage Toeplitz tile Btt[n][k] = K[(n0+n) - (kb*32+k)] while the DMA flies.
        {
            const int dbase = n0 + sn - kb * 32 - sk0;
            float* dst = &Btt[sn][sk0];
#pragma unroll
            for (int j = 0; j < 16; ++j) {
                const int d  = dbase - j;
                const int di = d < 0 ? 0 : d;
                float v = Kls[di];
                if (d < 0) v = 0.0f;
                dst[j] = v;
            }
        }
#if USE_TDM
        __builtin_amdgcn_s_wait_tensorcnt(0);
#endif
        __syncthreads();

#pragma unroll
        for (int kk = 0; kk < 8; ++kk) {
            const int kcol = kk * 4 + 2 * lhi;  // K index of VGPR0 in A/B frags
            v2f a[4];
#pragma unroll
            for (int im = 0; im < 4; ++im)
                a[im] = *(const v2f*)&Yt[wm * 64 + im * 16 + llo][kcol];
            v2f b[2];
#pragma unroll
            for (int in = 0; in < 2; ++in)
                b[in] = *(const v2f*)&Btt[wn * 32 + in * 16 + llo][kcol];
#pragma unroll
            for (int im = 0; im < 4; ++im)
#pragma unroll
                for (int in = 0; in < 2; ++in)
                    acc[im][in] = __builtin_amdgcn_wmma_f32_16x16x4_f32(
                        false, a[im], false, b[in], (short)0, acc[im][in], false, false);
        }
        __syncthreads();
    }

    const float Dv = Dp[0];
#pragma unroll
    for (int im = 0; im < 4; ++im) {
        const int mbase = m0 + wm * 64 + im * 16 + 8 * lhi;
#pragma unroll
        for (int in = 0; in < 2; ++in) {
            const int t = n0 + wn * 32 + in * 16 + llo;
#pragma unroll
            for (int v = 0; v < 8; ++v) {
                const size_t idx = (size_t)(mbase + v) * LSEQ + t;
                out[idx] = acc[im][in][v] + Dv * y[idx];
            }
        }
    }
}

// ---------------- host launcher ----------------

extern "C" void kernel_launch(void* const* d_in, const int* in_sizes, int n_in,
                              void* d_out, int out_size, void* d_ws, size_t ws_size,
                              hipStream_t stream) {
    const float* y  = (const float*)d_in[0];   // (2048, 4096)
    const float* A  = (const float*)d_in[1];   // (256, 256)
    const float* Bv = (const float*)d_in[2];   // (256, 1)
    const float* Cv = (const float*)d_in[3];   // (1, 256)
    const float* Dp = (const float*)d_in[4];   // (1,)
    float* out = (float*)d_out;
    float* ws  = (float*)d_ws;

    float* E    = ws;                 // 65536
    float* T1   = ws + 1 * 65536;     // 65536
    float* T2   = ws + 2 * 65536;     // 65536
    float* Abar = ws + 3 * 65536;     // 65536
    float* P1   = ws + 4 * 65536;     // 65536
    float* P2   = ws + 5 * 65536;     // 65536
    float* Bbar = ws + 6 * 65536;     // 256
    float* Kvec = Bbar + 256;         // 4096
    float* U    = Kvec + 4096;        // 256 * 4096

    const float stepv = 1.0f / (float)LSEQ;
    const float h     = 0.5f * stepv;

    dim3 b16(16, 16);

    // Phase 1: E = h*A ; Abar = I + 2(E+E^2+E^3+E^4) (Neumann/Horner); Bbar likewise
    k_scaleA<<<256, 256, 0, stream>>>(E, A, h);
    k_initT <<<256, 256, 0, stream>>>(T1, E);                          // I + E
    k_gemm256<<<dim3(16, 16), b16, 0, stream>>>(T2, E, T1, 256, 256, 256, 1);
    k_gemm256<<<dim3(16, 16), b16, 0, stream>>>(T1, E, T2, 256, 256, 256, 1);
    k_gemm256<<<dim3(16, 16), b16, 0, stream>>>(T2, E, T1, 256, 256, 256, 1);
    k_abar  <<<256, 256, 0, stream>>>(Abar, T2);
    k_bbar  <<<1, 256, 0, stream>>>(Bbar, E, Bv, stepv, 4);

    // Phase 2: log-doubling scan for U[:,l] = Abar^l Bbar, then K = C @ U
    k_setU0<<<1, 256, 0, stream>>>(U, Bbar);
    float* Pcur = Abar;
    float* Pnext = P1;
    for (int s = 0; s < 12; ++s) {
        const int k = 1 << s;
        k_gemm256<<<dim3((k + 15) / 16, 16), b16, 0, stream>>>(U + k, Pcur, U, k, LSEQ, LSEQ, 0);
        if (s < 11) {
            k_gemm256<<<dim3(16, 16), b16, 0, stream>>>(Pnext, Pcur, Pcur, 256, 256, 256, 0);
            Pcur = Pnext;
            Pnext = (Pcur == P1) ? P2 : P1;
        }
    }
    k_kvec<<<16, 256, 0, stream>>>(Kvec, Cv, U);

    // Phase 3: causal conv as f32 WMMA GEMM against the Toeplitz kernel matrix
    k_conv<<<dim3(32, 16), 256, 0, stream>>>(out, y, Kvec, Dp);
}